// SoftclDiceLoss_22969485099448
// MI455X (gfx1250) — compile-verified
//
#include <hip/hip_runtime.h>
#include <hip/hip_bf16.h>
#include <stdint.h>

// Problem geometry (2 batches x 2 arrays -> 4 independent volumes)
#define Wd 192
#define Hd 192
#define Dz 96
#define SLICE (Wd*Hd)
#define VOL (Dz*SLICE)          // 3,538,944
#define NVOL 4
#define NTOT (NVOL*VOL)         // 14,155,776

// Fused-open tile: output 16x16x8, erode halo radius 1 (18x18x10),
// input halo radius 2 (20x20x12).
#define TX 16
#define TY 16
#define TZ 8
#define E1X (TX+2)
#define E1Y (TY+2)
#define E1Z (TZ+2)
#define E1N (E1X*E1Y*E1Z)       // 3240 floats (12.96 KB)
#define BSX (TX+4)
#define BSY (TY+4)
#define BSZ (TZ+4)
#define BSN (BSX*BSY*BSZ)       // 4800 floats (19.2 KB)

#define RBLOCKS 1024
#define RTHREADS 256

#define NEG_INF (-3.402823466e+38f)
#define POS_INF ( 3.402823466e+38f)

typedef __attribute__((ext_vector_type(2))) float v2f;
typedef __attribute__((ext_vector_type(8))) float v8f;

// ---------------- CDNA5 async global -> LDS staging ----------------
__device__ __forceinline__ void async_g2lds_b32(uint32_t lds_byte_addr, const float* gptr) {
  // GV mode: per-lane LDS byte address in vdst, per-lane 64-bit global address
  asm volatile("global_load_async_to_lds_b32 %0, %1, off"
               :: "v"(lds_byte_addr), "v"(gptr)
               : "memory");
}
__device__ __forceinline__ void wait_asynccnt0() {
  asm volatile("s_wait_asynccnt 0" ::: "memory");
}

// ---------------- wave32 sum via V_WMMA_F32_16X16X4_F32 ----------------
__device__ __forceinline__ float wave_sum32(float v) {
#if __has_builtin(__builtin_amdgcn_wmma_f32_16x16x4_f32)
  v2f a; a[0] = v; a[1] = 0.0f;
  v2f b; b[0] = 1.0f; b[1] = 1.0f;
  v8f c = {};
  c = __builtin_amdgcn_wmma_f32_16x16x4_f32(false, a, false, b,
                                            (short)0, c, false, false);
  float t = c[0]+c[1]+c[2]+c[3]+c[4]+c[5]+c[6]+c[7];
  return __shfl(t, 0, 32) + __shfl(t, 16, 32);
#else
  for (int o = 16; o > 0; o >>= 1) v += __shfl_xor(v, o, 32);
  return v;
#endif
}

// ---------------- kernels ----------------

// Gather fg channel of both inputs into one 4-volume buffer (float4 path).
__global__ void k_slice(const float4* __restrict__ pred, const float4* __restrict__ tgt,
                        float4* __restrict__ img) {
  int i4 = blockIdx.x*blockDim.x + threadIdx.x;
  if (i4 >= NTOT/4) return;
  const int V4 = VOL/4;
  int vol = i4 / V4;
  int rem = i4 - vol*V4;
  float4 v = (vol < 2) ? pred[(size_t)(vol*2 + 1)*V4 + rem]
                       : tgt[(size_t)((vol-2)*2 + 1)*V4 + rem];
  img[i4] = v;
}

// 7-point cross min (soft_erode), 4 outputs/thread along W; OOB ignored.
__global__ void __launch_bounds__(256)
k_erode(const float* __restrict__ src, float* __restrict__ dst) {
  int i4 = blockIdx.x*blockDim.x + threadIdx.x;
  if (i4 >= NTOT/4) return;
  int base = i4 * 4;
  int x4 = base % Wd;
  int t  = base / Wd;
  int y  = t % Hd;
  int z  = (t / Hd) % Dz;

  float4 c = *(const float4*)(src + base);
  float l = (x4 > 0)       ? src[base-1] : POS_INF;
  float r = (x4 < Wd-4)    ? src[base+4] : POS_INF;

  float4 m;
  m.x = fminf(l,   fminf(c.x, c.y));
  m.y = fminf(c.x, fminf(c.y, c.z));
  m.z = fminf(c.y, fminf(c.z, c.w));
  m.w = fminf(c.z, fminf(c.w, r));

  if (y > 0) {
    float4 n = *(const float4*)(src + base - Wd);
    m.x = fminf(m.x, n.x); m.y = fminf(m.y, n.y);
    m.z = fminf(m.z, n.z); m.w = fminf(m.w, n.w);
  }
  if (y < Hd-1) {
    float4 n = *(const float4*)(src + base + Wd);
    m.x = fminf(m.x, n.x); m.y = fminf(m.y, n.y);
    m.z = fminf(m.z, n.z); m.w = fminf(m.w, n.w);
  }
  if (z > 0) {
    float4 n = *(const float4*)(src + base - SLICE);
    m.x = fminf(m.x, n.x); m.y = fminf(m.y, n.y);
    m.z = fminf(m.z, n.z); m.w = fminf(m.w, n.w);
  }
  if (z < Dz-1) {
    float4 n = *(const float4*)(src + base + SLICE);
    m.x = fminf(m.x, n.x); m.y = fminf(m.y, n.y);
    m.z = fminf(m.z, n.z); m.w = fminf(m.w, n.w);
  }
  *(float4*)(dst + base) = m;
}

// Fused soft_open + skel update:
//   E = erode(src) computed in LDS from a radius-2 halo of src,
//   O = dilate3x3x3(E), delta = relu(src - O),
//   INIT: skel = delta; else skel += relu(delta - skel*delta).
// Halo staged via async global->LDS loads (ASYNCcnt), imgRef read from LDS.
template<bool INIT>
__global__ void __launch_bounds__(TX*TY)
k_open_fuse(const float* __restrict__ src, float* __restrict__ skel) {
  __shared__ float bufS[BSN];   // src halo, radius 2; global-OOB = +inf
  __shared__ float e1[E1N];     // erode(src), radius 1; global-OOB = -inf

  const int tx  = threadIdx.x, ty = threadIdx.y;
  const int tid = ty*TX + tx;
  const int bx  = blockIdx.x*TX, by = blockIdx.y*TY;
  const int vol = blockIdx.z / (Dz/TZ);
  const int z0  = (blockIdx.z % (Dz/TZ))*TZ;
  const float* vbase = src + (size_t)vol*VOL;

  // Stage src halo (radius 2) into LDS with async loads.
  for (int li = tid; li < BSN; li += TX*TY) {
    int hx = li % BSX;
    int rr = li / BSX;
    int hy = rr % BSY;
    int hz = rr / BSY;
    int gx = bx - 2 + hx;
    int gy = by - 2 + hy;
    int gz = z0 - 2 + hz;
    bool ok = (gx >= 0) & (gx < Wd) & (gy >= 0) & (gy < Hd) & (gz >= 0) & (gz < Dz);
    if (ok) {
      uint32_t lds_b = (uint32_t)(uintptr_t)(&bufS[li]);
      const float* g = vbase + (size_t)gz*SLICE + (size_t)gy*Wd + gx;
      async_g2lds_b32(lds_b, g);
    } else {
      bufS[li] = POS_INF;            // min identity: OOB ignored by erode
    }
  }
  wait_asynccnt0();
  __syncthreads();

  // Erode (7-point cross min) into e1; cells outside the volume are -inf
  // so the subsequent dilate-max ignores them (reduce_window -inf padding).
  for (int li = tid; li < E1N; li += TX*TY) {
    int hx = li % E1X;
    int rr = li / E1X;
    int hy = rr % E1Y;
    int hz = rr / E1Y;
    int gx = bx - 1 + hx;
    int gy = by - 1 + hy;
    int gz = z0 - 1 + hz;
    bool ok = (gx >= 0) & (gx < Wd) & (gy >= 0) & (gy < Hd) & (gz >= 0) & (gz < Dz);
    float v = NEG_INF;
    if (ok) {
      int cz = hz + 1, cy = hy + 1, cx = hx + 1;   // bufS coords
      #define BS(a,b,cc) bufS[((a)*BSY + (b))*BSX + (cc)]
      v = BS(cz, cy, cx);
      v = fminf(v, BS(cz, cy, cx-1));
      v = fminf(v, BS(cz, cy, cx+1));
      v = fminf(v, BS(cz, cy-1, cx));
      v = fminf(v, BS(cz, cy+1, cx));
      v = fminf(v, BS(cz-1, cy, cx));
      v = fminf(v, BS(cz+1, cy, cx));
      #undef BS
    }
    e1[li] = v;
  }
  __syncthreads();

  // Dilate from e1 + fused skel update; imgRef comes from bufS center (LDS).
  for (int z = 0; z < TZ; ++z) {
    float m = NEG_INF;
#pragma unroll
    for (int dz = 0; dz < 3; ++dz)
#pragma unroll
      for (int dy = 0; dy < 3; ++dy)
#pragma unroll
        for (int dx = 0; dx < 3; ++dx)
          m = fmaxf(m, e1[((z+dz)*E1Y + (ty+dy))*E1X + (tx+dx)]);

    float iref  = bufS[((z+2)*BSY + (ty+2))*BSX + (tx+2)];
    float delta = fmaxf(iref - m, 0.0f);
    size_t gidx = ((size_t)(vol*Dz + z0 + z)*Hd + (by+ty))*(size_t)Wd + (bx+tx);
    if (INIT) {
      skel[gidx] = delta;
    } else {
      float s = skel[gidx];
      skel[gidx] = s + fmaxf(delta - s*delta, 0.0f);
    }
  }
}

// Deterministic partial reduction: per-block (num_p, den_p, num_t, den_t).
__global__ void __launch_bounds__(RTHREADS)
k_reduce(const float* __restrict__ skel,
         const float* __restrict__ pred,
         const float* __restrict__ tgt,
         float* __restrict__ partials) {
  float a0 = 0.f, a1 = 0.f, a2 = 0.f, a3 = 0.f;
  for (int idx = blockIdx.x*RTHREADS + threadIdx.x; idx < NTOT; idx += RBLOCKS*RTHREADS) {
    int vol = idx / VOL;
    int rem = idx - vol*VOL;
    float s = skel[idx];
    if (vol < 2) {                       // skel_pred vs target_fg
      float w = tgt[(size_t)(vol*2 + 1)*VOL + rem];
      a0 += s*w; a1 += s;
    } else {                             // skel_target vs pred_fg
      float w = pred[(size_t)((vol-2)*2 + 1)*VOL + rem];
      a2 += s*w; a3 += s;
    }
  }
  float r0 = wave_sum32(a0);
  float r1 = wave_sum32(a1);
  float r2 = wave_sum32(a2);
  float r3 = wave_sum32(a3);

  __shared__ float ws4[RTHREADS/32][4];
  int wid  = threadIdx.x >> 5;
  int lane = threadIdx.x & 31;
  if (lane == 0) { ws4[wid][0]=r0; ws4[wid][1]=r1; ws4[wid][2]=r2; ws4[wid][3]=r3; }
  __syncthreads();
  if (threadIdx.x == 0) {
    float t0=0.f, t1=0.f, t2=0.f, t3=0.f;
    for (int w = 0; w < RTHREADS/32; ++w) {
      t0 += ws4[w][0]; t1 += ws4[w][1]; t2 += ws4[w][2]; t3 += ws4[w][3];
    }
    partials[blockIdx.x*4+0] = t0;
    partials[blockIdx.x*4+1] = t1;
    partials[blockIdx.x*4+2] = t2;
    partials[blockIdx.x*4+3] = t3;
  }
}

__global__ void k_final(const float* __restrict__ partials, float* __restrict__ out) {
  if (blockIdx.x == 0 && threadIdx.x == 0) {
    float n0=0.f, d0=0.f, n1=0.f, d1=0.f;
    for (int i = 0; i < RBLOCKS; ++i) {
      n0 += partials[i*4+0]; d0 += partials[i*4+1];
      n1 += partials[i*4+2]; d1 += partials[i*4+3];
    }
    const float S = 1e-5f;
    float tprec = (n0 + S)/(d0 + S);
    float tsens = (n1 + S)/(d1 + S);
    float cl    = 2.0f*tprec*tsens/(tprec + tsens + S);
    out[0] = 1.0f - cl;
  }
}

// ---------------- host-side pipeline ----------------
extern "C" void kernel_launch(void* const* d_in, const int* in_sizes, int n_in,
                              void* d_out, int out_size, void* d_ws, size_t ws_size,
                              hipStream_t stream) {
  const float* pred = (const float*)d_in[0];
  const float* tgt  = (const float*)d_in[1];
  float* ws = (float*)d_ws;

  float* A = ws;                          // current img (4 volumes)
  float* B = ws + (size_t)NTOT;           // eroded img
  float* S = ws + (size_t)2*NTOT;         // skel
  float* partials = ws + (size_t)3*NTOT;  // 4*RBLOCKS floats
  // Working set A+B+S = 170 MB -> resident in the 192 MB L2.

  dim3 vb(256);
  dim3 vg((NTOT/4 + 255)/256);
  dim3 db(TX, TY, 1);
  dim3 dg(Wd/TX, Hd/TY, (Dz/TZ)*NVOL);

  // skel = relu(img - dilate(erode(img)))  -- fully fused
  k_slice<<<vg, vb, 0, stream>>>((const float4*)pred, (const float4*)tgt, (float4*)A);
  k_open_fuse<true><<<dg, db, 0, stream>>>(A, S);

  // 50 iterations: img' = erode(img); delta = relu(img' - open(img')); skel update
  for (int it = 0; it < 50; ++it) {
    k_erode<<<vg, vb, 0, stream>>>(A, B);          // img' = erode(img)
    k_open_fuse<false><<<dg, db, 0, stream>>>(B, S);
    float* t = A; A = B; B = t;                    // img = img'
  }

  k_reduce<<<RBLOCKS, RTHREADS, 0, stream>>>(S, pred, tgt, partials);
  k_final<<<1, 64, 0, stream>>>(partials, (float*)d_out);
}